// TropicalAttention_23295902613799
// MI455X (gfx1250) — compile-verified
//
#include <hip/hip_runtime.h>
#include <hip/hip_bf16.h>

// Tropical top-k attention for MI455X (gfx1250, wave32).
// B=2, T=512, d=256, H=4, dh=64, TOP_K=8, causal.
//
// Design (compile-only reasoning, see analysis):
//  - All four projections are FP32 WMMA GEMMs (V_WMMA_F32_16X16X4_F32):
//    top-k index selection is precision
//    sensitive, so we keep full fp32 instead of f16/bf16 WMMA.
//  - Tropical (max-plus) scoring is a semiring op -> VALU (v_add/v_max),
//    one wave per query row, b128 loads of K rows, Q row staged in LDS.
//  - Top-8 via per-lane partial maxima + wave32 butterfly argmax.
// Entire working set (~5 MB) is L2-resident (192 MB); problem is
// latency-bound, so kernels are kept simple and wide.

#define TT   512
#define DD   256
#define HH   4
#define DHH  64
#define MTOT 1024   // B*T rows

typedef float v2f __attribute__((ext_vector_type(2)));
typedef float v8f __attribute__((ext_vector_type(8)));

// ---------------------------------------------------------------------------
// C(M=1024, N=256) = A(1024x256) @ W(256x256)^T   (y[m,n] = sum_k A[m,k]W[n,k])
// One wave computes a 16x16 tile; K-loop in steps of 4 with fp32 WMMA.
// FP32 fragment layouts (ISA 7.12.2):
//   A 16x4 : lanes 0-15 row M=lane, VGPR{0,1}=K{0,1}; lanes 16-31 K{2,3}
//   B 4x16 : lanes 0-15 col N=lane, VGPR{0,1}=K{0,1}; lanes 16-31 K{2,3}
//   C 16x16: VGPR g -> row g (lanes 0-15) / row g+8 (lanes 16-31), col=lane&15
// Both A and B fragments are contiguous float2 loads here.
// ---------------------------------------------------------------------------
__global__ void __launch_bounds__(128)
gemm_xWT_wmma_f32(const float* __restrict__ A, const float* __restrict__ W,
                  float* __restrict__ C) {
    const int lane = threadIdx.x & 31;
    const int wave = threadIdx.x >> 5;
    const int m0 = (blockIdx.x * 4 + wave) * 16;  // 64 m-tiles
    const int n0 = blockIdx.y * 16;               // 16 n-tiles
    const int l15   = lane & 15;
    const int khalf = (lane >> 4) << 1;           // 0 or 2

    const float* ap = A + (m0 + l15) * DD + khalf;
    const float* bp = W + (n0 + l15) * DD + khalf;

    v8f acc = {};
    #pragma unroll 8
    for (int kk = 0; kk < DD; kk += 4) {
        __builtin_prefetch(bp + kk + 64, 0, 0);   // global_prefetch_b8 on W stream
        v2f a = *(const v2f*)(ap + kk);
        v2f b = *(const v2f*)(bp + kk);
        acc = __builtin_amdgcn_wmma_f32_16x16x4_f32(
            /*neg_a=*/false, a, /*neg_b=*/false, b,
            /*c_mod=*/(short)0, acc, /*reuse_a=*/false, /*reuse_b=*/false);
    }

    const int rbase = m0 + ((lane >> 4) << 3);
    #pragma unroll
    for (int g = 0; g < 8; ++g)
        C[(rbase + g) * DD + n0 + l15] = acc[g];
}

// ---------------------------------------------------------------------------
// Tropical max-plus scores + causal mask + top-8 + softmax + V-gather.
// One wave per query (b,h,i); 4 waves per block. Q,K,V are (B*T, 256)
// row-major; head h occupies columns [h*64, h*64+64).
// ---------------------------------------------------------------------------
__global__ void __launch_bounds__(128)
tropical_topk_attn(const float* __restrict__ Q, const float* __restrict__ K,
                   const float* __restrict__ V, float* __restrict__ ctx) {
    __shared__ __align__(16) float qs[4][DHH];

    const int lane = threadIdx.x & 31;
    const int wave = threadIdx.x >> 5;
    const int flat = blockIdx.x * 4 + wave;   // ((b*H + h)*T + i), 0..4095
    const int i  = flat & (TT - 1);
    const int bh = flat >> 9;
    const int h  = bh & (HH - 1);
    const int b  = bh >> 2;
    const int rowbase = b * TT * DD + h * DHH;   // + t*DD selects token t

    // Stage Q row (64 floats) in LDS for broadcast reads.
    const float* qrow = Q + rowbase + i * DD;
    qs[wave][lane]      = qrow[lane];
    qs[wave][lane + 32] = qrow[lane + 32];
    __syncthreads();

    // Each lane scores keys j = lane + 32*c, c in [0,16); causal j <= i.
    float s[16];
    #pragma unroll
    for (int c = 0; c < 16; ++c) s[c] = -__builtin_inff();

    const float4* q4 = (const float4*)qs[wave];
    for (int c = 0; c < 16; ++c) {
        const int j = lane + (c << 5);
        if (j <= i) {
            const float4* k4 = (const float4*)(K + rowbase + j * DD);
            float sm = -__builtin_inff();
            #pragma unroll 4
            for (int d4 = 0; d4 < 16; ++d4) {
                float4 kv = k4[d4];
                float4 qv = q4[d4];
                sm = fmaxf(sm, qv.x + kv.x);
                sm = fmaxf(sm, qv.y + kv.y);
                sm = fmaxf(sm, qv.z + kv.z);
                sm = fmaxf(sm, qv.w + kv.w);
            }
            s[c] = sm;
        }
    }

    // Top-8 extraction: per-lane scan + wave32 butterfly argmax, 8 rounds.
    float topv[8];
    int   topj[8];
    #pragma unroll
    for (int k = 0; k < 8; ++k) {
        float bv = s[0];
        int   bc = 0;
        #pragma unroll
        for (int c = 1; c < 16; ++c)
            if (s[c] > bv) { bv = s[c]; bc = c; }
        int bj = lane + (bc << 5);
        #pragma unroll
        for (int off = 16; off > 0; off >>= 1) {
            float ov = __shfl_xor(bv, off, 32);
            int   oj = __shfl_xor(bj, off, 32);
            if (ov > bv || (ov == bv && oj < bj)) { bv = ov; bj = oj; }
        }
        topv[k] = bv;
        topj[k] = bj;
        if ((bj & 31) == lane) s[bj >> 5] = -__builtin_inff();  // retire winner
    }

    // Softmax over the top-8 (topv[0] is the row max; -inf slots -> weight 0).
    const float m = topv[0];
    float w[8], wsum = 0.f;
    #pragma unroll
    for (int k = 0; k < 8; ++k) { w[k] = __expf(topv[k] - m); wsum += w[k]; }
    const float inv = 1.0f / wsum;

    // ctx[d] = sum_k attn[k] * V[j_k][d]; lane owns d = lane and d = lane+32.
    float acc0 = 0.f, acc1 = 0.f;
    #pragma unroll
    for (int k = 0; k < 8; ++k) {
        const float  a    = w[k] * inv;
        const float* vrow = V + rowbase + topj[k] * DD;
        acc0 += a * vrow[lane];
        acc1 += a * vrow[lane + 32];
    }
    float* crow = ctx + rowbase + i * DD;
    crow[lane]      = acc0;
    crow[lane + 32] = acc1;
}

// ---------------------------------------------------------------------------
// Launch: 3x projection GEMM -> attention -> output GEMM.
// Workspace: Q,K,V,ctx = 4 * 1 MB fp32.
// ---------------------------------------------------------------------------
extern "C" void kernel_launch(void* const* d_in, const int* in_sizes, int n_in,
                              void* d_out, int out_size, void* d_ws, size_t ws_size,
                              hipStream_t stream) {
    const float* x  = (const float*)d_in[0];
    const float* Wq = (const float*)d_in[1];
    const float* Wk = (const float*)d_in[2];
    const float* Wv = (const float*)d_in[3];
    const float* Wo = (const float*)d_in[4];
    float* out = (float*)d_out;

    float* Qw = (float*)d_ws;
    float* Kw = Qw + MTOT * DD;
    float* Vw = Kw + MTOT * DD;
    float* Cw = Vw + MTOT * DD;   // needs 4 MB of workspace

    dim3 ggrid(16, 16);           // 64 m-tiles (4 waves/block) x 16 n-tiles
    dim3 gblk(128);

    gemm_xWT_wmma_f32<<<ggrid, gblk, 0, stream>>>(x, Wq, Qw);
    gemm_xWT_wmma_f32<<<ggrid, gblk, 0, stream>>>(x, Wk, Kw);
    gemm_xWT_wmma_f32<<<ggrid, gblk, 0, stream>>>(x, Wv, Vw);

    tropical_topk_attn<<<dim3(1024), gblk, 0, stream>>>(Qw, Kw, Vw, Cw);

    gemm_xWT_wmma_f32<<<ggrid, gblk, 0, stream>>>(Cw, Wo, out);
}